// Taskselector_1477468750023
// MI455X (gfx1250) — compile-verified
//
#include <hip/hip_runtime.h>
#include <math.h>

typedef __attribute__((ext_vector_type(2))) float v2f;
typedef __attribute__((ext_vector_type(4))) float v4f;
typedef __attribute__((ext_vector_type(8))) float v8f;

#define HDIM 300
#define WROW (2 * HDIM)                                    // 600 columns per W row
#define ROWS_PER_WAVE 16
#define WAVES_PER_BLOCK 8
#define ROWS_PER_BLOCK (ROWS_PER_WAVE * WAVES_PER_BLOCK)   // 128
#define BLOCK_THREADS (WAVES_PER_BLOCK * 32)               // 256, wave32
#define EPSF 1e-20f

__global__ __launch_bounds__(BLOCK_THREADS)
void task_selector_kernel(const float* __restrict__ se1,
                          const float* __restrict__ se2,
                          const float* __restrict__ Wg,   // [2][600]
                          const float* __restrict__ bg,   // [2]
                          const float* __restrict__ ug,   // [B][2]
                          float* __restrict__ outg,       // [B][600]
                          int Bn) {
    // Row 0,1: W rows. Row 2: zeros, fed to B-fragment lanes N>=2 so no
    // per-iteration masking is needed in the WMMA loop.
    __shared__ float ldsW[3 * WROW];                              // 1800 floats
    __shared__ float ldsSel[WAVES_PER_BLOCK][ROWS_PER_WAVE * 2];  // logits -> selectors

    const int tid  = threadIdx.x;
    const int lane = tid & 31;
    const int wave = tid >> 5;

    // stage W (4.8 KB) + a zeroed dummy row into LDS once per block
    for (int i = tid; i < 2 * WROW; i += BLOCK_THREADS)
        ldsW[i] = Wg[i];
    for (int i = tid; i < WROW; i += BLOCK_THREADS)
        ldsW[2 * WROW + i] = 0.0f;
    __syncthreads();

    const int r0 = blockIdx.x * ROWS_PER_BLOCK + wave * ROWS_PER_WAVE;
    const bool active = (r0 + ROWS_PER_WAVE) <= Bn;   // wave-uniform

    // ISA lane layout for 32-bit A (16x4) and B (4x16):
    //   lanes 0-15 hold K = k+0 / k+1 (VGPR0/1), lanes 16-31 hold K = k+2 / k+3
    const int n     = lane & 15;   // A row (M) for this lane / B column (N)
    const int half  = lane >> 4;
    const int khalf = half * 2;
    const int arow  = r0 + n;

    // ---------------- Phase 1: logits = se @ W^T via V_WMMA_F32_16X16X4_F32 ----
    if (active) {
        v8f acc = {0.f,0.f,0.f,0.f,0.f,0.f,0.f,0.f};
        const int wsel = (n < 2) ? n : 2;              // lanes N>=2 read the zero row
        const float* wbase = &ldsW[wsel * WROW + khalf];

        // first 300 K-values come from se1 (W columns 0..299)
        {
            const float* abase = se1 + (size_t)arow * HDIM + khalf;
            for (int k = 0; k < HDIM; k += 4) {
                v2f a  = *(const v2f*)(abase + k);
                v2f bm = *(const v2f*)(wbase + k);
                acc = __builtin_amdgcn_wmma_f32_16x16x4_f32(
                        false, a, false, bm, (short)0, acc, false, false);
            }
        }
        // next 300 K-values come from se2 (W columns 300..599)
        {
            const float* abase  = se2 + (size_t)arow * HDIM + khalf;
            const float* wbase2 = wbase + HDIM;
            for (int k = 0; k < HDIM; k += 4) {
                v2f a  = *(const v2f*)(abase + k);
                v2f bm = *(const v2f*)(wbase2 + k);
                acc = __builtin_amdgcn_wmma_f32_16x16x4_f32(
                        false, a, false, bm, (short)0, acc, false, false);
            }
        }

        // D tile layout: VGPR j, lanes 0-15 -> D[M=j][N=lane]; lanes 16-31 -> D[M=8+j][N=lane-16]
        if (n < 2) {
            const int mbase = half * 8;
            float* dst = &ldsSel[wave][0];
            #pragma unroll
            for (int j = 0; j < 8; ++j)
                dst[(mbase + j) * 2 + n] = acc[j];
        }
    }
    __syncthreads();

    // ---------------- Phase 2: per-row relu + log_softmax + gumbel ST selector --
    if (active && lane < ROWS_PER_WAVE) {
        const int m   = lane;
        const int row = r0 + m;
        float z0 = ldsSel[wave][m*2+0] + bg[0];
        float z1 = ldsSel[wave][m*2+1] + bg[1];
        z0 = fmaxf(z0, 0.0f);
        z1 = fmaxf(z1, 0.0f);
        // log_softmax
        float mx  = fmaxf(z0, z1);
        float lse = mx + logf(expf(z0 - mx) + expf(z1 - mx));
        float l0 = z0 - lse, l1 = z1 - lse;
        // gumbel noise from uniform
        float u0 = ug[(size_t)row*2 + 0];
        float u1 = ug[(size_t)row*2 + 1];
        float g0 = -logf(-logf(u0 + EPSF) + EPSF);
        float g1 = -logf(-logf(u1 + EPSF) + EPSF);
        // soft probs
        float s0 = l0 + g0, s1 = l1 + g1;
        float smx = fmaxf(s0, s1);
        float e0 = expf(s0 - smx), e1 = expf(s1 - smx);
        float inv = 1.0f / (e0 + e1);
        float y0 = e0 * inv, y1 = e1 * inv;
        // argmax (ties -> index 0, matching jnp.argmax) + straight-through estimator
        float h0 = (y0 >= y1) ? 1.0f : 0.0f;
        float h1 = 1.0f - h0;
        ldsSel[wave][m*2+0] = (h0 - y0) + y0;
        ldsSel[wave][m*2+1] = (h1 - y1) + y1;
    }
    __syncthreads();

    // ---------------- Phase 3: streamed out = se * selector (float4, coalesced) --
    if (active) {
        for (int m = 0; m < ROWS_PER_WAVE; ++m) {
            const int row   = r0 + m;
            const float s0  = ldsSel[wave][m*2+0];
            const float s1  = ldsSel[wave][m*2+1];
            const v4f* a1 = (const v4f*)(se1 + (size_t)row * HDIM);
            const v4f* a2 = (const v4f*)(se2 + (size_t)row * HDIM);
            v4f* o1 = (v4f*)(outg + (size_t)row * 2 * HDIM);
            v4f* o2 = o1 + (HDIM / 4);
            for (int c = lane; c < HDIM / 4; c += 32) {
                o1[c] = a1[c] * s0;
                o2[c] = a2[c] * s1;
            }
        }
    }
}

extern "C" void kernel_launch(void* const* d_in, const int* in_sizes, int n_in,
                              void* d_out, int out_size, void* d_ws, size_t ws_size,
                              hipStream_t stream) {
    const float* se1 = (const float*)d_in[0];   // [B,300]
    const float* se2 = (const float*)d_in[1];   // [B,300]
    const float* W   = (const float*)d_in[2];   // [2,600]
    const float* b   = (const float*)d_in[3];   // [2]
    const float* u   = (const float*)d_in[4];   // [B,2]
    float* out = (float*)d_out;                 // [B,600]
    const int Bn = in_sizes[0] / HDIM;
    const int grid = (Bn + ROWS_PER_BLOCK - 1) / ROWS_PER_BLOCK;
    hipLaunchKernelGGL(task_selector_kernel, dim3(grid), dim3(BLOCK_THREADS), 0, stream,
                       se1, se2, W, b, u, out, Bn);
}